// InvRadonLayer_29746943492402
// MI455X (gfx1250) — compile-verified
//
#include <hip/hip_runtime.h>

typedef __attribute__((ext_vector_type(2))) float v2f;
typedef __attribute__((ext_vector_type(8))) float v8f;

#define H_IN    512
#define W_IN    360
#define D_OUT   512
#define N_BATCH 8
#define WTILES  23   // ceil(360/16)

// ---------------------------------------------------------------------------
// Kernel A: build circulant filter matrix  M[y][m] = h[(m - y - 257) mod 512]
// (rf[n,y,w] = sum_k h[k] * radon[n,(y+257+k) mod 512, w]  ==  M @ radon)
// ---------------------------------------------------------------------------
__global__ void build_filter_matrix(const float* __restrict__ h,
                                    float* __restrict__ M) {
  int idx = blockIdx.x * blockDim.x + threadIdx.x;   // 0 .. 512*512-1
  int y = idx >> 9;
  int m = idx & 511;
  // (m - y - 257) mod 512 == (m - y + 767) & 511   (always non-negative)
  M[idx] = h[(m - y + 767) & 511];
}

// ---------------------------------------------------------------------------
// Kernel B: rf[n,y,w] = sum_k M[y,k] * radon[n,k,w]   via V_WMMA_F32_16X16X4_F32
// One wave (32 lanes) per 16x16 output tile; K=512 in steps of 4.
// Output layout: rf[(w*8 + n)*512 + y]   (column-major in y, per (w,n) column)
//
// Fragment layouts per CDNA5 ISA 7.12.2:
//   A (16x4 f32):  row M = lane%16 ; VGPR e holds K = 2*(lane/16) + e
//   B (4x16 f32):  col N = lane%16 ; VGPR e holds K = 2*(lane/16) + e
//   C/D (16x16):   VGPR j: lanes 0-15 -> M=j, lanes 16-31 -> M=j+8, N=lane%16
// ---------------------------------------------------------------------------
__global__ void filter_gemm_wmma(const float* __restrict__ M,
                                 const float* __restrict__ radon,
                                 float* __restrict__ rf) {
  const int lane  = threadIdx.x;        // 0..31, one wave per block
  const int ytile = blockIdx.x * 16;    // 0..511
  const int wtile = blockIdx.y * 16;    // 0..352 (last tile partial)
  const int n     = blockIdx.z;         // batch
  const int half  = lane >> 4;          // 0 or 1
  const int l15   = lane & 15;

  const int yrow = ytile + l15;                     // A-fragment row
  const int wreq = wtile + l15;                     // requested B column
  const int wcol = wreq < W_IN ? wreq : W_IN - 1;   // clamped (keep EXEC full)

  const float* __restrict__ Arow = M + (size_t)yrow * H_IN;
  const float* __restrict__ Bcol = radon + (size_t)n * H_IN * W_IN + wcol;

  v8f c = {};
  for (int k0 = 0; k0 < H_IN; k0 += 4) {
    const int kk = k0 + 2 * half;
    v2f a, b;
    a[0] = Arow[kk];
    a[1] = Arow[kk + 1];
    b[0] = Bcol[(size_t)kk * W_IN];
    b[1] = Bcol[(size_t)(kk + 1) * W_IN];
    // 8 args: (neg_a, A, neg_b, B, c_mod, C, reuse_a, reuse_b)
    c = __builtin_amdgcn_wmma_f32_16x16x4_f32(false, a, false, b,
                                              (short)0, c, false, false);
  }

  // Lane holds rf[y = ytile + 8*half + j][w = wreq] in c[j], j = 0..7:
  // 8 consecutive y values -> two contiguous float4 stores.
  if (wreq < W_IN) {
    float* colp = rf + ((size_t)(wreq * N_BATCH + n) * H_IN + ytile + 8 * half);
    float4* p = reinterpret_cast<float4*>(colp);   // 32B aligned
    p[0] = make_float4(c[0], c[1], c[2], c[3]);
    p[1] = make_float4(c[4], c[5], c[6], c[7]);
  }
}

// ---------------------------------------------------------------------------
// Kernel C: backprojection. Each thread owns one pixel p=(r,c), keeps 8 batch
// accumulators in registers, loops over 360 angles. tG read once, coalesced.
// rf gathers hit L2/WGP$ (active footprint per angle: 2 cols x 8 batches=32KB).
// ---------------------------------------------------------------------------
__global__ void backproject(const float* __restrict__ tG,
                            const float* __restrict__ rf,
                            float* __restrict__ out) {
  const int p = blockIdx.x * blockDim.x + threadIdx.x;   // 0 .. 512*512-1

  float acc[N_BATCH];
#pragma unroll
  for (int n = 0; n < N_BATCH; ++n) acc[n] = 0.0f;

  const float2* __restrict__ g2 = reinterpret_cast<const float2*>(tG);
  const size_t plane = (size_t)D_OUT * D_OUT;

  for (int i = 0; i < W_IN; ++i) {
    const float2 g = g2[(size_t)i * plane + p];
    if (i + 1 < W_IN)  // gfx1250 global_prefetch_b8 of next angle's grid line
      __builtin_prefetch(&g2[(size_t)(i + 1) * plane + p], 0, 0);

    const float gx = (g.x + 1.0f) * 0.5f * (float)(W_IN - 1);
    const float gy = (g.y + 1.0f) * 0.5f * (float)(H_IN - 1);
    const float x0f = floorf(gx), y0f = floorf(gy);
    const float wx1 = gx - x0f,  wy1 = gy - y0f;
    const float wx0 = 1.0f - wx1, wy0 = 1.0f - wy1;
    const int x0 = (int)x0f, y0 = (int)y0f;
    const int x1 = x0 + 1,   y1 = y0 + 1;

    const float vx0 = (x0 >= 0 && x0 < W_IN) ? 1.0f : 0.0f;
    const float vx1 = (x1 >= 0 && x1 < W_IN) ? 1.0f : 0.0f;
    const float vy0 = (y0 >= 0 && y0 < H_IN) ? 1.0f : 0.0f;
    const float vy1 = (y1 >= 0 && y1 < H_IN) ? 1.0f : 0.0f;

    const float w00 = wx0 * wy0 * vx0 * vy0;
    const float w10 = wx1 * wy0 * vx1 * vy0;
    const float w01 = wx0 * wy1 * vx0 * vy1;
    const float w11 = wx1 * wy1 * vx1 * vy1;

    const int xc0 = min(max(x0, 0), W_IN - 1);
    const int xc1 = min(max(x1, 0), W_IN - 1);
    const int yc0 = min(max(y0, 0), H_IN - 1);
    const int yc1 = min(max(y1, 0), H_IN - 1);

    const float* __restrict__ base0 = rf + (size_t)(xc0 * N_BATCH) * H_IN;
    const float* __restrict__ base1 = rf + (size_t)(xc1 * N_BATCH) * H_IN;
#pragma unroll
    for (int n = 0; n < N_BATCH; ++n) {
      const float* c0 = base0 + n * H_IN;
      const float* c1 = base1 + n * H_IN;
      acc[n] += w00 * c0[yc0] + w01 * c0[yc1] +
                w10 * c1[yc0] + w11 * c1[yc1];
    }
  }

  const float scale = 3.14159265358979323846f / (2.0f * (float)W_IN);
#pragma unroll
  for (int n = 0; n < N_BATCH; ++n)
    out[(size_t)n * plane + p] = acc[n] * scale;
}

// ---------------------------------------------------------------------------
// Host-side launcher (graph-capture safe: async launches on `stream` only)
// ---------------------------------------------------------------------------
extern "C" void kernel_launch(void* const* d_in, const int* in_sizes, int n_in,
                              void* d_out, int out_size, void* d_ws, size_t ws_size,
                              hipStream_t stream) {
  const float* radon = (const float*)d_in[0];  // (8,1,512,360) f32
  const float* hG    = (const float*)d_in[1];  // (1,1,512,1)   f32
  const float* tG    = (const float*)d_in[2];  // (360,512,512,2) f32
  float* out = (float*)d_out;                  // (8,1,512,512) f32

  float* M  = (float*)d_ws;                    // 512*512 f32  = 1 MB
  float* rf = M + (size_t)H_IN * H_IN;         // 360*8*512 f32 ~ 5.9 MB

  // A) build circulant ramp-filter matrix
  build_filter_matrix<<<(H_IN * H_IN) / 256, 256, 0, stream>>>(hG, M);

  // B) filter GEMM: rf[n] = M (512x512) @ radon[n] (512x360), fp32 WMMA
  dim3 gridB(H_IN / 16, WTILES, N_BATCH);
  filter_gemm_wmma<<<gridB, 32, 0, stream>>>(M, radon, rf);

  // C) backprojection over 360 angles
  backproject<<<(D_OUT * D_OUT) / 256, 256, 0, stream>>>(tG, rf, out);
}